// Predictor_82205674045928
// MI455X (gfx1250) — compile-verified
//
#include <hip/hip_runtime.h>
#include <hip/hip_bf16.h>

// CDNA5 (gfx1250) GraphSAGE-GCN encoder + MLP decoder.
// Roofline: scatter-add of 230M fp32 atomics (~400MB traffic) dominates;
// GEMMs are only 16.4 GFLOP -> use exact fp32 WMMA (V_WMMA_F32_16X16X4_F32),
// wave32, 16x16 tiles, LDS-staged normalized A tiles.

typedef float v2f __attribute__((ext_vector_type(2)));
typedef float v8f __attribute__((ext_vector_type(8)));

#define TM    16        // node rows per block / WMMA M,N
#define MAXK  256       // max K (HID)
#define LDPAD 2         // even pad: conflict-free banks + 8B-aligned K pairs

__device__ __forceinline__ void atomAddF(float* p, float v) {
  __hip_atomic_fetch_add(p, v, __ATOMIC_RELAXED, __HIP_MEMORY_SCOPE_AGENT);
}

__global__ void zero_f32(float* __restrict__ p, long long n) {
  long long i = (long long)blockIdx.x * blockDim.x + threadIdx.x;
  long long stride = (long long)gridDim.x * blockDim.x;
  for (; i < n; i += stride) p[i] = 0.0f;
}

__global__ void degree_count(const int* __restrict__ dst, float* __restrict__ deg, int E) {
  int i = blockIdx.x * blockDim.x + threadIdx.x;
  int stride = gridDim.x * blockDim.x;
  for (; i < E; i += stride) atomAddF(&deg[dst[i]], 1.0f);
}

// One block handles (256 / (F/4)) edges; each lane moves one float4 chunk.
// Consecutive lanes -> consecutive float4 -> fully coalesced gather.
__global__ void scatter_add4(const float* __restrict__ X, const int* __restrict__ src,
                             const int* __restrict__ dst, float* __restrict__ agg,
                             int E, int F) {
  const int cpe = F >> 2;                 // float4 chunks per edge (32 or 64)
  const int epb = 256 / cpe;              // edges per block
  const int e = blockIdx.x * epb + (int)threadIdx.x / cpe;
  if (e >= E) return;
  const int ch = ((int)threadIdx.x % cpe) << 2;
  const int s = src[e], d = dst[e];
  const float4 v = *reinterpret_cast<const float4*>(X + (size_t)s * F + ch);
  float* p = agg + (size_t)d * F + ch;
  atomAddF(p + 0, v.x); atomAddF(p + 1, v.y);
  atomAddF(p + 2, v.z); atomAddF(p + 3, v.w);
}

// fp32 WMMA K-loop over a staged LDS A tile and global B (weights).
// A frag: lane holds (row=lane&15, K=kb..kb+1), kb = 2*(lane>>4)  [ISA 7.12.2]
// B frag: lane holds (col=lane&15, same K pair)
__device__ __forceinline__ v8f wmma_kloop(const float* __restrict__ As, int lda,
                                          const float* __restrict__ W, int Hout,
                                          int K, int row, int kb, int col) {
  v8f acc = {};
  for (int k0 = 0; k0 < K; k0 += 4) {
    v2f a;
    a.x = As[row * lda + k0 + kb];
    a.y = As[row * lda + k0 + kb + 1];
    v2f b;
    b.x = W[(size_t)(k0 + kb) * Hout + col];
    b.y = W[(size_t)(k0 + kb + 1) * Hout + col];
    acc = __builtin_amdgcn_wmma_f32_16x16x4_f32(false, a, false, b, (short)0, acc,
                                                false, false);
  }
  return acc;
}

// Y[N,Hout] = act( ((AGG + X) / (deg+1)) @ W + bias ).  256 thr = 8 waves,
// block covers 16 nodes x all Hout columns (each wave: 2 col tiles).
__global__ __launch_bounds__(256)
void sage_gemm(const float* __restrict__ X, const float* __restrict__ AGG,
               const float* __restrict__ DEG, const float* __restrict__ W,
               const float* __restrict__ bias, float* __restrict__ Y,
               int K, int Hout, int relu) {
  __shared__ float As[TM * (MAXK + LDPAD)];
  __shared__ float rs[TM];
  const int lda = K + LDPAD;
  const int base = blockIdx.x * TM;
  const int tid = threadIdx.x;

  if (tid < TM) rs[tid] = 1.0f / (DEG[base + tid] + 1.0f);
  __syncthreads();
  for (int idx = tid; idx < TM * K; idx += 256) {
    const int r = idx / K, c = idx - r * K;
    const size_t g = (size_t)(base + r) * K + c;
    As[r * lda + c] = (X[g] + AGG[g]) * rs[r];
  }
  __syncthreads();

  const int lane = tid & 31;
  const int row = lane & 15;
  const int kb = (lane >> 4) << 1;
  const int nct = Hout >> 4;
  for (int ct = tid >> 5; ct < nct; ct += 8) {
    const int col = (ct << 4) + (lane & 15);
    v8f acc = wmma_kloop(As, lda, W, Hout, K, row, kb, col);
    const float bv = bias[col];
#pragma unroll
    for (int v = 0; v < 8; ++v) {
      float val = acc[v] + bv;                       // D row = 8*(lane>>4)+v
      if (relu) val = fmaxf(val, 0.0f);
      Y[(size_t)(base + ((lane >> 4) << 3) + v) * Hout + col] = val;
    }
  }
}

// Fused: h2 = norm(h1) @ W2 + b2 ; h3 = relu(h2 @ Wd1 + bd1) ; out = h3 @ Wd2 + bd2
// Keeps h2/h3 tiles entirely in LDS (saves two 51MB HBM round trips).
__global__ __launch_bounds__(256)
void sage2_decoder(const float* __restrict__ H1, const float* __restrict__ AGG,
                   const float* __restrict__ DEG,
                   const float* __restrict__ W2, const float* __restrict__ b2,
                   const float* __restrict__ Wd1, const float* __restrict__ bd1,
                   const float* __restrict__ Wd2, const float* __restrict__ bd2,
                   float* __restrict__ out, int H) {
  __shared__ float As[TM * (MAXK + LDPAD)];   // norm input tile, reused for h3
  __shared__ float Hs[TM * (MAXK + LDPAD)];   // h2 tile
  __shared__ float red[TM][17];
  __shared__ float rs[TM];
  const int lda = H + LDPAD;
  const int base = blockIdx.x * TM;
  const int tid = threadIdx.x;

  if (tid < TM) rs[tid] = 1.0f / (DEG[base + tid] + 1.0f);
  __syncthreads();
  for (int idx = tid; idx < TM * H; idx += 256) {
    const int r = idx / H, c = idx - r * H;
    const size_t g = (size_t)(base + r) * H + c;
    As[r * lda + c] = (H1[g] + AGG[g]) * rs[r];
  }
  __syncthreads();

  const int lane = tid & 31;
  const int row = lane & 15;
  const int kb = (lane >> 4) << 1;
  const int nct = H >> 4;

  // layer-2 GEMM -> Hs (no relu)
  for (int ct = tid >> 5; ct < nct; ct += 8) {
    const int col = (ct << 4) + (lane & 15);
    v8f acc = wmma_kloop(As, lda, W2, H, H, row, kb, col);
    const float bv = b2[col];
#pragma unroll
    for (int v = 0; v < 8; ++v)
      Hs[(((lane >> 4) << 3) + v) * lda + col] = acc[v] + bv;
  }
  __syncthreads();

  // decoder linear1 + relu -> As (reused as h3 tile)
  for (int ct = tid >> 5; ct < nct; ct += 8) {
    const int col = (ct << 4) + (lane & 15);
    v8f acc = wmma_kloop(Hs, lda, Wd1, H, H, row, kb, col);
    const float bv = bd1[col];
#pragma unroll
    for (int v = 0; v < 8; ++v)
      As[(((lane >> 4) << 3) + v) * lda + col] = fmaxf(acc[v] + bv, 0.0f);
  }
  __syncthreads();

  // decoder linear2: out[r] = h3[r,:] . Wd2 + bd2
  {
    const int r = tid >> 4, q = tid & 15;
    const int seg = H >> 4;                      // 16 elements per partial
    float s = 0.0f;
    for (int c = q * seg; c < (q + 1) * seg; ++c)
      s += As[r * lda + c] * Wd2[c];
    red[r][q] = s;
  }
  __syncthreads();
  if (tid < TM) {
    float t = 0.0f;
#pragma unroll
    for (int q = 0; q < 16; ++q) t += red[tid][q];
    out[base + tid] = t + bd2[0];
  }
}

extern "C" void kernel_launch(void* const* d_in, const int* in_sizes, int n_in,
                              void* d_out, int out_size, void* d_ws, size_t ws_size,
                              hipStream_t stream) {
  const float* feats = (const float*)d_in[0];
  const int*   src   = (const int*)d_in[1];
  const int*   dst   = (const int*)d_in[2];
  const float* W1    = (const float*)d_in[3];
  const float* b1    = (const float*)d_in[4];
  const float* W2    = (const float*)d_in[5];
  const float* b2    = (const float*)d_in[6];
  const float* Wd1   = (const float*)d_in[7];
  const float* bd1   = (const float*)d_in[8];
  const float* Wd2   = (const float*)d_in[9];
  const float* bd2   = (const float*)d_in[10];
  float* out = (float*)d_out;

  const int E = in_sizes[1];
  const int H = in_sizes[4];                  // HID  (256)
  const int F = in_sizes[3] / H;              // FEAT (128)
  const int N = in_sizes[0] / F;              // 50000 (multiple of 16)

  float* wsf = (float*)d_ws;
  const long long degPad = (long long)((N + 255) & ~255);
  float* deg = wsf;
  float* agg = wsf + degPad;                  // N*F for layer1, reused N*H for layer2
  float* h1  = agg + (long long)N * H;
  const long long need = degPad + 2LL * N * H;
  if (ws_size < (size_t)need * sizeof(float)) return;

  // layer 1: zero(deg+agg1) -> degree -> scatter -> norm-GEMM (relu)
  zero_f32<<<2048, 256, 0, stream>>>(wsf, degPad + (long long)N * F);
  degree_count<<<1024, 256, 0, stream>>>(dst, deg, E);
  {
    const int epb = 256 / (F >> 2);
    scatter_add4<<<(E + epb - 1) / epb, 256, 0, stream>>>(feats, src, dst, agg, E, F);
  }
  sage_gemm<<<N / TM, 256, 0, stream>>>(feats, agg, deg, W1, b1, h1, F, H, 1);

  // layer 2 + decoder: re-zero agg (now N*H) -> scatter h1 -> fused kernel
  zero_f32<<<2048, 256, 0, stream>>>(agg, (long long)N * H);
  {
    const int epb = 256 / (H >> 2);
    scatter_add4<<<(E + epb - 1) / epb, 256, 0, stream>>>(h1, src, dst, agg, E, H);
  }
  sage2_decoder<<<N / TM, 256, 0, stream>>>(h1, agg, deg, W2, b2, Wd1, bd1, Wd2, bd2,
                                            out, H);
}